// MultiHeadDeformableAttention2D_38354057954012
// MI455X (gfx1250) — compile-verified
//
#include <hip/hip_runtime.h>

typedef __attribute__((ext_vector_type(2))) float v2f;
typedef __attribute__((ext_vector_type(8))) float v8f;

#define NHEADS  8
#define NPOINTS 4
#define E_DIM   256
#define L_DIM   1024      // 32*32
#define HW      32
#define NROWS   4096      // N * L

// -------------------------------------------------------------------------
// Kernel 1: fused projection GEMMs + softmax + sampling-location table.
// grid = 256 blocks (one per 16-row tile of the 4096 flattened query rows)
// block = 192 threads = 6 waves; wave t owns one 16x16 column tile of the
// fused projection [offs(64 cols) | attn(32 cols)], K=256 via
// V_WMMA_F32_16X16X4_F32.
// -------------------------------------------------------------------------
__global__ __launch_bounds__(192)
void k1_proj_sampling(const float* __restrict__ q,       // (4096, 256)
                      const float* __restrict__ refpts,  // (4096, 2)
                      const float* __restrict__ Woff,    // (256, 64)
                      const float* __restrict__ boff,    // (64)
                      const float* __restrict__ Wattn,   // (256, 32)
                      const float* __restrict__ battn,   // (32)
                      float4* __restrict__ samp)         // (4096, 32)
{
  __shared__ float proj[16][97];                  // 96 cols + pad
  const int tid  = threadIdx.x;
  const int wave = tid >> 5;
  const int lane = tid & 31;
  const int half = lane >> 4;                     // 0: lanes 0-15, 1: 16-31
  const int l15  = lane & 15;
  const int rbase = blockIdx.x * 16;

  // A operand: row = rbase + l15, K pair offset = half*2
  const float* Ap = q + (size_t)(rbase + l15) * E_DIM + half * 2;

  const float* Wb; int ncols, cb;
  if (wave < 4) { Wb = Woff;  ncols = NHEADS * NPOINTS * 2; cb = wave * 16; }
  else          { Wb = Wattn; ncols = NHEADS * NPOINTS;     cb = (wave - 4) * 16; }
  const float* Bp = Wb + (size_t)(half * 2) * ncols + cb + l15;

  v8f c = {};
  for (int k = 0; k < E_DIM; k += 4) {
    v2f a, b;
    a.x = Ap[k];
    a.y = Ap[k + 1];
    b.x = Bp[(size_t)k * ncols];
    b.y = Bp[(size_t)(k + 1) * ncols];
    c = __builtin_amdgcn_wmma_f32_16x16x4_f32(false, a, false, b,
                                              (short)0, c, false, false);
  }

  const float bias = (wave < 4) ? boff[cb + l15] : battn[cb + l15];
#pragma unroll
  for (int r = 0; r < 8; ++r)
    proj[r + half * 8][wave * 16 + l15] = c[r] + bias;

  __syncthreads();

  // Epilogue: 16 rows x 32 (h,p) = 512 sample records over 192 threads.
  for (int i = tid; i < 16 * NHEADS * NPOINTS; i += 192) {
    const int r  = i >> 5;
    const int hp = i & 31;
    const int h  = hp >> 2;
    const int p  = hp & 3;

    // softmax over the 4 points of head h
    const float a0 = proj[r][64 + h * 4 + 0];
    const float a1 = proj[r][64 + h * 4 + 1];
    const float a2 = proj[r][64 + h * 4 + 2];
    const float a3 = proj[r][64 + h * 4 + 3];
    const float m  = fmaxf(fmaxf(a0, a1), fmaxf(a2, a3));
    const float e0 = expf(a0 - m), e1 = expf(a1 - m);
    const float e2 = expf(a2 - m), e3 = expf(a3 - m);
    const float den = e0 + e1 + e2 + e3;
    const float ep  = (p == 0) ? e0 : (p == 1) ? e1 : (p == 2) ? e2 : e3;
    const float aw  = ep / (den * (float)NHEADS);   // attn / NUM_HEADS

    const float ox = proj[r][(h * NPOINTS + p) * 2 + 0];
    const float oy = proj[r][(h * NPOINTS + p) * 2 + 1];

    const int   g  = rbase + r;                     // flattened (n,l)
    const float rx = refpts[2 * g + 0];
    const float ry = refpts[2 * g + 1];
    // ((ref+off)*2-1 + 1) * size * 0.5 - 0.5 == (ref+off)*size - 0.5
    const float x  = (rx + ox) * (float)HW - 0.5f;
    const float y  = (ry + oy) * (float)HW - 0.5f;
    const float xf = floorf(x), yf = floorf(y);
    const int   x0 = (int)xf,   y0 = (int)yf;

    float4 s;
    s.x = __int_as_float((x0 & 0xffff) | (y0 << 16));
    s.y = x - xf;
    s.z = y - yf;
    s.w = aw;
    samp[(size_t)g * 32 + hp] = s;
  }
}

// -------------------------------------------------------------------------
// Kernel 2: bilinear gather + weighted accumulation over (head, point).
// grid = 4096 blocks (one per (n,l)); block = 256 threads (one channel each).
// Gather rows are 1 KB contiguous -> fully coalesced; query slices are
// L2-resident.
// -------------------------------------------------------------------------
__global__ __launch_bounds__(256)
void k2_gather(const float* __restrict__ q,       // (N, L, E)
               const float4* __restrict__ samp,   // (4096, 32)
               float* __restrict__ mid)           // (4096, 256)
{
  __shared__ float4 s_s[32];
  const int g = blockIdx.x;
  const int e = threadIdx.x;
  if (e < 32) s_s[e] = samp[(size_t)g * 32 + e];
  __syncthreads();

  const float* qn = q + (size_t)(g >> 10) * L_DIM * E_DIM;  // batch slice
  float acc = 0.f;
#pragma unroll 4
  for (int s = 0; s < 32; ++s) {
    const float4 sp = s_s[s];
    const int xy = __float_as_int(sp.x);
    const int x0 = (int)(short)(xy & 0xffff);
    const int y0 = xy >> 16;
    const float wx = sp.y, wy = sp.z, aw = sp.w;
    const float w00 = aw * (1.f - wx) * (1.f - wy);
    const float w10 = aw * wx * (1.f - wy);
    const float w01 = aw * (1.f - wx) * wy;
    const float w11 = aw * wx * wy;
    const int x1 = x0 + 1, y1 = y0 + 1;
    if ((unsigned)x0 < 32u && (unsigned)y0 < 32u)
      acc += w00 * qn[(size_t)(y0 * HW + x0) * E_DIM + e];
    if ((unsigned)x1 < 32u && (unsigned)y0 < 32u)
      acc += w10 * qn[(size_t)(y0 * HW + x1) * E_DIM + e];
    if ((unsigned)x0 < 32u && (unsigned)y1 < 32u)
      acc += w01 * qn[(size_t)(y1 * HW + x0) * E_DIM + e];
    if ((unsigned)x1 < 32u && (unsigned)y1 < 32u)
      acc += w11 * qn[(size_t)(y1 * HW + x1) * E_DIM + e];
  }
  mid[(size_t)g * E_DIM + e] = acc;
}

// -------------------------------------------------------------------------
// Kernel 3: out = mid @ W_out + b_out  (4096 x 256 x 256), fp32 WMMA.
// grid = 512 blocks x 256 threads (8 waves); wave owns one 16x16 tile.
// All 8 waves of a block share the same column tile (B reuse via L0).
// -------------------------------------------------------------------------
__global__ __launch_bounds__(256)
void k3_out_gemm(const float* __restrict__ mid,   // (4096, 256)
                 const float* __restrict__ Wout,  // (256, 256)
                 const float* __restrict__ bout,  // (256)
                 float* __restrict__ out)         // (4096, 256)
{
  const int lane = threadIdx.x & 31;
  const int wave = threadIdx.x >> 5;
  const int half = lane >> 4;
  const int l15  = lane & 15;
  const int colTile = blockIdx.x & 15;
  const int rowTile = (blockIdx.x >> 4) * 8 + wave;
  const int r0 = rowTile * 16;
  const int c0 = colTile * 16;

  const float* Ap = mid  + (size_t)(r0 + l15) * E_DIM + half * 2;
  const float* Bp = Wout + (size_t)(half * 2) * E_DIM + c0 + l15;

  v8f c = {};
  for (int k = 0; k < E_DIM; k += 4) {
    v2f a, b;
    a.x = Ap[k];
    a.y = Ap[k + 1];
    b.x = Bp[(size_t)k * E_DIM];
    b.y = Bp[(size_t)(k + 1) * E_DIM];
    c = __builtin_amdgcn_wmma_f32_16x16x4_f32(false, a, false, b,
                                              (short)0, c, false, false);
  }

  const float bias = bout[c0 + l15];
#pragma unroll
  for (int r = 0; r < 8; ++r) {
    const int row = r0 + r + half * 8;
    out[(size_t)row * E_DIM + c0 + l15] = c[r] + bias;
  }
}

// -------------------------------------------------------------------------
extern "C" void kernel_launch(void* const* d_in, const int* in_sizes, int n_in,
                              void* d_out, int out_size, void* d_ws, size_t ws_size,
                              hipStream_t stream) {
  (void)in_sizes; (void)n_in; (void)out_size; (void)ws_size;
  const float* q     = (const float*)d_in[0];
  const float* refp  = (const float*)d_in[1];
  const float* Woff  = (const float*)d_in[2];
  const float* boff  = (const float*)d_in[3];
  const float* Wattn = (const float*)d_in[4];
  const float* battn = (const float*)d_in[5];
  const float* Wout  = (const float*)d_in[6];
  const float* bout  = (const float*)d_in[7];
  float* out = (float*)d_out;

  float4* samp = (float4*)d_ws;                                   // 2 MB
  float*  mid  = (float*)((char*)d_ws +
                          (size_t)NROWS * 32 * sizeof(float4));   // 4 MB

  k1_proj_sampling<<<NROWS / 16, 192, 0, stream>>>(q, refp, Woff, boff,
                                                   Wattn, battn, samp);
  k2_gather<<<NROWS, 256, 0, stream>>>(q, samp, mid);
  k3_out_gemm<<<NROWS / 8, 256, 0, stream>>>(mid, Wout, bout, out);
}